// crop_layer_65695819759919
// MI455X (gfx1250) — compile-verified
//
#include <hip/hip_runtime.h>
#include <stdint.h>

#define BATCH 32
#define HH 512
#define WW 512
#define OUTD 400
#define THRESH 0.7f
#define SLICES 32                       // mask slices per batch (pass A)

// ---- CDNA5 async global->LDS helpers (ASYNCcnt path) ----------------------
// s_wait_dscnt 0 first: guarantees any prior ds_load of this buffer has fully
// completed (data in VGPRs) before the overwriting async DMA is requested.
// (DScnt and ASYNCcnt are independent counters; this closes the WAR window.)
__device__ __forceinline__ void async_b128(uint64_t sbase, uint32_t voff_bytes,
                                           uint32_t lds_addr) {
    asm volatile("s_wait_dscnt 0" ::: "memory");
    // GVS mode: mem = SGPR64 + VGPR32 offset; VDST = per-lane LDS byte addr.
    asm volatile("global_load_async_to_lds_b128 %0, %1, %2 offset:0"
                 :: "v"(lds_addr), "v"(voff_bytes), "s"(sbase) : "memory");
}
#define ASYNC_WAIT_1() asm volatile("s_wait_asynccnt 1" ::: "memory")
#define ASYNC_WAIT_0() asm volatile("s_wait_asynccnt 0" ::: "memory")

// Branchless bbox accumulation over 4 staged floats (one row, 4 | 512).
__device__ __forceinline__ void accum4(const float* buf, int lin,
                                       int& rmin, int& rmax, int& cmin, int& cmax) {
    const float4 v = *(const float4*)buf;          // ds_load_b128
    const int r  = lin >> 9;
    const int cb = lin & (WW - 1);
    const float e[4] = {v.x, v.y, v.z, v.w};
#pragma unroll
    for (int l = 0; l < 4; ++l) {
        const bool h = e[l] > THRESH;
        rmin = min(rmin, h ? r      : 0x7fffffff);
        rmax = max(rmax, h ? r      : -1);
        cmin = min(cmin, h ? cb + l : 0x7fffffff);
        cmax = max(cmax, h ? cb + l : -1);
    }
}

// ---------------------------------------------------------------------------
// Kernel 0: sentinel-init the 32 bbox records (re-run every call: stateless).
// ---------------------------------------------------------------------------
__global__ void init_bounds(int* __restrict__ bounds) {
    const int t = threadIdx.x;
    if (t < BATCH * 4) bounds[t] = (t & 1) ? -1 : 0x7fffffff;  // {min,max,min,max}
}

// ---------------------------------------------------------------------------
// Kernel A: partial bbox per (slice, batch); combined via 4 global atomics.
// Mask slice (32 KB) streamed via double-buffered async global->LDS copies.
// Each lane stages 16 B into its own LDS slot => s_wait_asynccnt alone
// (no barrier) orders producer->consumer; async loads complete in order.
// ---------------------------------------------------------------------------
__global__ __launch_bounds__(256) void bounds_kernel(const float* __restrict__ tensor,
                                                     int* __restrict__ bounds) {
    const int b   = blockIdx.y;
    const int s   = blockIdx.x;
    const int tid = threadIdx.x;

    __shared__ float stage[2][1024];               // 8 KB double buffer
    __shared__ int   s_red[4];                     // rmin,rmax,cmin,cmax

    const int SLICE_FLOATS = (HH * WW) / SLICES;   // 8192
    const int base_lin     = s * SLICE_FLOATS;
    const uint64_t sbase =
        (uint64_t)(uintptr_t)(tensor + (size_t)b * (HH * WW) + base_lin);

    // Low 32 bits of a flat shared pointer == LDS logical byte address.
    const uint32_t lds0 = (uint32_t)(uintptr_t)&stage[0][tid * 4];
    const uint32_t lds1 = (uint32_t)(uintptr_t)&stage[1][tid * 4];

    int rmin = 0x7fffffff, rmax = -1, cmin = 0x7fffffff, cmax = -1;

    const int NCH = SLICE_FLOATS / 1024;           // 8 chunks of 4 KB
    async_b128(sbase, (uint32_t)(tid * 16), lds0);
#pragma unroll
    for (int ch = 0; ch < NCH; ch += 2) {
        if (ch + 1 < NCH) { async_b128(sbase, (uint32_t)((ch + 1) * 4096 + tid * 16), lds1); ASYNC_WAIT_1(); }
        else              { ASYNC_WAIT_0(); }
        accum4(&stage[0][tid * 4], base_lin + ch * 1024 + tid * 4, rmin, rmax, cmin, cmax);

        if (ch + 2 < NCH) { async_b128(sbase, (uint32_t)((ch + 2) * 4096 + tid * 16), lds0); ASYNC_WAIT_1(); }
        else              { ASYNC_WAIT_0(); }
        accum4(&stage[1][tid * 4], base_lin + (ch + 1) * 1024 + tid * 4, rmin, rmax, cmin, cmax);
    }

    // wave32 butterfly reduce
#pragma unroll
    for (int m = 16; m >= 1; m >>= 1) {
        rmin = min(rmin, __shfl_xor(rmin, m, 32));
        rmax = max(rmax, __shfl_xor(rmax, m, 32));
        cmin = min(cmin, __shfl_xor(cmin, m, 32));
        cmax = max(cmax, __shfl_xor(cmax, m, 32));
    }
    if (tid < 4) s_red[tid] = (tid & 1) ? -1 : 0x7fffffff;
    __syncthreads();
    if ((tid & 31) == 0) {
        atomicMin(&s_red[0], rmin); atomicMax(&s_red[1], rmax);
        atomicMin(&s_red[2], cmin); atomicMax(&s_red[3], cmax);
    }
    __syncthreads();
    if (tid == 0) {
        atomicMin(&bounds[b * 4 + 0], s_red[0]);
        atomicMax(&bounds[b * 4 + 1], s_red[1]);
        atomicMin(&bounds[b * 4 + 2], s_red[2]);
        atomicMax(&bounds[b * 4 + 3], s_red[3]);
    }
}

// ---------------------------------------------------------------------------
// Kernel B: bilinear crop-resize. One thread = 4 consecutive output pixels in
// one row (4 | 400; 4 px * 12 B = 48 B, 16-B aligned => 3x b128 stores).
// grid = (ceil(40000/256), 32) so bounds reads are block-uniform (scalar).
// ---------------------------------------------------------------------------
__global__ __launch_bounds__(256) void resize_kernel(const float* __restrict__ img,
                                                     const int* __restrict__ bounds,
                                                     float* __restrict__ out) {
    const int b    = blockIdx.y;
    const int quad = blockIdx.x * 256 + threadIdx.x;
    if (quad >= (OUTD * OUTD) / 4) return;
    const int pix = quad * 4;
    const int i   = pix / OUTD;          // output row (same for all 4 pixels)
    const int j   = pix - i * OUTD;      // first output col

    const int4 bb = *(const int4*)(bounds + b * 4);
    int rlo = bb.x, rhi = bb.y, clo = bb.z, chi = bb.w;
    if (rhi < 0) { rlo = 0; rhi = HH - 1; }     // all-false mask: argmax semantics
    if (chi < 0) { clo = 0; chi = WW - 1; }

    // Row coords (JAX: src=(i+0.5)*size/OUT-0.5; clip[0,max(size-1,0)];
    //             i1=min(i0+1, max(hi-lo-1,0)))
    const float sr   = (float)(rhi - rlo);
    float srcr = ((float)i + 0.5f) * sr / (float)OUTD - 0.5f;
    srcr = fminf(fmaxf(srcr, 0.0f), fmaxf(sr - 1.0f, 0.0f));
    const int   i0 = (int)floorf(srcr);
    const int   i1 = min(i0 + 1, max(rhi - rlo - 1, 0));
    const float wr = srcr - (float)i0;
    const float omwr = 1.0f - wr;

    const float* row0 = img + ((size_t)b * HH + (rlo + i0)) * (WW * 3);
    const float* row1 = img + ((size_t)b * HH + (rlo + i1)) * (WW * 3);

    const float sc      = (float)(chi - clo);
    const float sc_hi   = fmaxf(sc - 1.0f, 0.0f);
    const int   j1_clmp = max(chi - clo - 1, 0);

    float res[12];
#pragma unroll
    for (int q = 0; q < 4; ++q) {
        float srcc = ((float)(j + q) + 0.5f) * sc / (float)OUTD - 0.5f;
        srcc = fminf(fmaxf(srcc, 0.0f), sc_hi);
        const int   j0 = (int)floorf(srcc);
        const int   j1 = min(j0 + 1, j1_clmp);
        const float wc = srcc - (float)j0;
        const float omwc = 1.0f - wc;
        const int c0 = (clo + j0) * 3;
        const int c1 = (clo + j1) * 3;
#pragma unroll
        for (int ch = 0; ch < 3; ++ch) {
            const float top = row0[c0 + ch] * omwc + row0[c1 + ch] * wc;
            const float bot = row1[c0 + ch] * omwc + row1[c1 + ch] * wc;
            res[q * 3 + ch] = top * omwr + bot * wr;
        }
    }

    float4* o = (float4*)(out + ((size_t)(b * OUTD + i) * OUTD + j) * 3);
    o[0] = make_float4(res[0], res[1], res[2],  res[3]);
    o[1] = make_float4(res[4], res[5], res[6],  res[7]);
    o[2] = make_float4(res[8], res[9], res[10], res[11]);
}

extern "C" void kernel_launch(void* const* d_in, const int* in_sizes, int n_in,
                              void* d_out, int out_size, void* d_ws, size_t ws_size,
                              hipStream_t stream) {
    const float* image  = (const float*)d_in[0];   // (32,512,512,3) f32
    const float* tensor = (const float*)d_in[1];   // (32,512,512,1) f32
    float* out  = (float*)d_out;                   // (32,400,400,3) f32
    int*   bnds = (int*)d_ws;                      // 32 * 4 ints scratch

    init_bounds<<<dim3(1), dim3(128), 0, stream>>>(bnds);
    bounds_kernel<<<dim3(SLICES, BATCH), dim3(256), 0, stream>>>(tensor, bnds);

    dim3 grid(((OUTD * OUTD) / 4 + 255) / 256, BATCH);
    resize_kernel<<<grid, dim3(256), 0, stream>>>(image, bnds, out);
}